// CMLILoss_6227702579749
// MI455X (gfx1250) — compile-verified
//
#include <hip/hip_runtime.h>
#include <math.h>

// Problem constants (from reference): B=64, N=197, M=65, D=768
#define B_    64
#define N_    197
#define M_    65
#define D_    768
#define NROWS 196              // image tokens after dropping CLS
#define MCOLS 64               // text tokens after dropping CLS
#define NTILE 13               // ceil(196/16) n-tiles of 16 rows
#define LDSW  65               // padded LDS row stride (bank-conflict-free column scans)

typedef __attribute__((ext_vector_type(2))) float v2f;
typedef __attribute__((ext_vector_type(8))) float v8f;

// One workgroup per (i,j) batch pair: 4 waves x 32 lanes.
// Wave w owns text-token columns [16w, 16w+16); each wave accumulates all 13
// 16x16 n-tiles (K=768 in steps of 4 via V_WMMA_F32_16X16X4_F32), writes the
// scaled 196x64 sim block to LDS, then the block reduces it.
__global__ void __launch_bounds__(128)
cmli_sim_kernel(const float* __restrict__ image,
                const float* __restrict__ text,
                const int*   __restrict__ pmask,
                const float* __restrict__ scale_p,
                float*       __restrict__ out)
{
    __shared__ float lds[NROWS * LDSW];   // ~51 KB of the 320 KB WGP LDS
    __shared__ float red[128];
    __shared__ int   pmsh[MCOLS];

    const int j    = blockIdx.x;          // text batch index
    const int i    = blockIdx.y;          // image batch index
    const int tid  = threadIdx.x;
    const int w    = tid >> 5;            // wave id 0..3  -> m-tile
    const int lane = tid & 31;
    const int mrow = lane & 15;           // row-in-tile / col-in-tile
    const int hi   = lane >> 4;           // lane half: K sub-pair selector

    const float* imgI = image + (size_t)i * N_ * D_ + D_;  // skip CLS token
    const float* txtJ = text  + (size_t)j * M_ * D_ + D_;

    // Padding mask row for image index i (reference broadcasts pm over i).
    if (tid < MCOLS) pmsh[tid] = pmask[i * M_ + 1 + tid];

    // B fragment source: one text row per lane (column-of-sim = text token).
    const int    col  = w * 16 + mrow;                 // 0..63
    const float* bRow = txtJ + (size_t)col * D_;

    // A fragment sources: one image row per lane per n-tile. Rows >=196 are
    // clamped (duplicate row 195) and excluded from all reductions below.
    const float* aRow[NTILE];
#pragma unroll
    for (int t = 0; t < NTILE; ++t) {
        int r = t * 16 + mrow;
        if (r > NROWS - 1) r = NROWS - 1;
        aRow[t] = imgI + (size_t)r * D_;
    }

    v8f c[NTILE];
    const v8f zero8 = {};
#pragma unroll
    for (int t = 0; t < NTILE; ++t) c[t] = zero8;

    // fp32 WMMA layout (16x16x4): lanes 0-15 hold K={k0,k0+1}, lanes 16-31
    // hold K={k0+2,k0+3} -> per-lane operand is a contiguous 8B pair.
    const int koff = 2 * hi;
    for (int k0 = 0; k0 < D_; k0 += 4) {
        const v2f b = *(const v2f*)(bRow + k0 + koff);   // reused by 13 tiles
#pragma unroll
        for (int t = 0; t < NTILE; ++t) {
            const v2f a = *(const v2f*)(aRow[t] + k0 + koff);
            c[t] = __builtin_amdgcn_wmma_f32_16x16x4_f32(
                /*neg_a=*/false, a, /*neg_b=*/false, b,
                /*c_mod=*/(short)0, c[t], /*reuse_a=*/false, /*reuse_b=*/false);
        }
    }

    // Scale BEFORE max (exact reference semantics, valid for any sign of s),
    // spill the 196x64 block to LDS. C/D layout: VGPR r -> row r + 8*hi.
    const float s = scale_p[0];
#pragma unroll
    for (int t = 0; t < NTILE; ++t) {
#pragma unroll
        for (int r = 0; r < 8; ++r) {
            const int row = t * 16 + hi * 8 + r;
            if (row < NROWS) lds[row * LDSW + col] = s * c[t][r];
        }
    }
    __syncthreads();

    // ---- logits_per_image[i,j] = mean_n max_{m valid} sim ----
    float sum = 0.f;
    for (int n = tid; n < NROWS; n += 128) {
        float mx = -INFINITY;
        const float* rowp = &lds[n * LDSW];
        for (int m = 0; m < MCOLS; ++m)
            if (pmsh[m] == 0) mx = fmaxf(mx, rowp[m]);
        sum += mx;
    }
    red[tid] = sum;
    __syncthreads();
    for (int st = 64; st > 0; st >>= 1) {
        if (tid < st) red[tid] += red[tid + st];
        __syncthreads();
    }
    if (tid == 0) out[1 + i * B_ + j] = red[0] / (float)NROWS;
    __syncthreads();

    // ---- logits_per_text[i,j] = sum_{m valid}(max_n sim) / cnt[i] ----
    float contrib = 0.f;
    if (tid < MCOLS) {
        float mx = -INFINITY;
        for (int n = 0; n < NROWS; ++n)
            mx = fmaxf(mx, lds[n * LDSW + tid]);       // stride-65: conflict-free
        if (pmsh[tid] == 0) contrib = mx;
    }
    red[tid] = contrib;
    __syncthreads();
    for (int st = 64; st > 0; st >>= 1) {
        if (tid < st) red[tid] += red[tid + st];
        __syncthreads();
    }
    if (tid == 0) {
        int cnt = 0;
        for (int m = 0; m < MCOLS; ++m) cnt += (pmsh[m] == 0);
        out[1 + B_ * B_ + i * B_ + j] = red[0] / (float)cnt;
    }
}

// Single block, 64 threads: diagonal log-softmax CE over both 64x64 logit
// matrices, final loss, and targets = arange(64).
__global__ void __launch_bounds__(64)
cmli_loss_kernel(float* __restrict__ out)
{
    __shared__ float red[64];
    const int i = threadIdx.x;

    const float* Li = out + 1 + i * B_;               // logits_per_image row i
    const float* Lt = out + 1 + B_ * B_ + i * B_;     // logits_per_text  row i

    float d = 0.f;
    for (int which = 0; which < 2; ++which) {
        const float* L = which ? Lt : Li;
        float mx = -INFINITY;
        for (int jj = 0; jj < B_; ++jj) mx = fmaxf(mx, L[jj]);
        float se = 0.f;
        for (int jj = 0; jj < B_; ++jj) se += expf(L[jj] - mx);
        d += L[i] - (logf(se) + mx);                  // diag(log_softmax)
    }
    red[i] = d;
    __syncthreads();
    for (int st = 32; st > 0; st >>= 1) {
        if (i < st) red[i] += red[i + st];
        __syncthreads();
    }
    if (i == 0) out[0] = -red[0] / (2.f * (float)B_); // 0.5*(ce_img+ce_text)
    out[1 + 2 * B_ * B_ + i] = (float)i;              // targets
}

extern "C" void kernel_launch(void* const* d_in, const int* in_sizes, int n_in,
                              void* d_out, int out_size, void* d_ws, size_t ws_size,
                              hipStream_t stream) {
    const float* image  = (const float*)d_in[0];   // (64,197,768) f32
    const float* text   = (const float*)d_in[1];   // (64,65,768)  f32
    const int*   pmask  = (const int*)d_in[2];     // (64,65)      i32
    const float* scale  = (const float*)d_in[3];   // ()           f32
    float*       out    = (float*)d_out;           // [loss | Limg 4096 | Ltxt 4096 | targets 64]

    dim3 grid(B_, B_);                             // 4096 workgroups, one per (i,j)
    cmli_sim_kernel<<<grid, 128, 0, stream>>>(image, text, pmask, scale, out);
    cmli_loss_kernel<<<1, 64, 0, stream>>>(out);
}